// GCN_25348896981056
// MI455X (gfx1250) — compile-verified
//
#include <hip/hip_runtime.h>
#include <math.h>

typedef __attribute__((ext_vector_type(2))) float v2f;
typedef __attribute__((ext_vector_type(8))) float v8f;

#define N_NODES  50000
#define N_EDGES  800000
#define N_GRAPHS 512
#define IN_F     30
#define KPAD     32
#define H1       256
#define H2       128
#define H3       64

static __device__ __forceinline__ void atomAddF(float* p, float v) {
    __hip_atomic_fetch_add(p, v, __ATOMIC_RELAXED, __HIP_MEMORY_SCOPE_AGENT);
}

// ---------------- degree / normalization ----------------
__global__ void k_deg_init(unsigned* __restrict__ deg) {
    int i = blockIdx.x * blockDim.x + threadIdx.x;
    if (i < N_NODES) deg[i] = 1u;                 // self-loop
}
__global__ void k_deg_count(const int* __restrict__ dst, unsigned* __restrict__ deg) {
    int i = blockIdx.x * blockDim.x + threadIdx.x;
    if (i < N_EDGES) atomicAdd(&deg[dst[i]], 1u);
}
__global__ void k_deg_inv(const unsigned* __restrict__ deg, float* __restrict__ dinv) {
    int i = blockIdx.x * blockDim.x + threadIdx.x;
    if (i < N_NODES) dinv[i] = rsqrtf((float)deg[i]);
}

// ---------------- pad x: [N,30] -> [N,32] with zero cols 30,31 ----------------
__global__ void k_pack_x(const float* __restrict__ x, float* __restrict__ xp) {
    int i = blockIdx.x * blockDim.x + threadIdx.x;
    if (i >= N_NODES * KPAD) return;
    const int n = i >> 5, c = i & 31;
    xp[i] = (c < IN_F) ? x[(size_t)n * IN_F + c] : 0.0f;
}

// ---------------- fp32 WMMA GEMM: Y[N,F] = X[N,K] @ W[K,F] ----------------
// compile-time K,F: guard-free inner loop, one wave per 16x16 output tile.
template<int K, int F>
__global__ void k_gemm_wmma(const float* __restrict__ X, const float* __restrict__ W,
                            float* __restrict__ Y) {
    const int lane = threadIdx.x & 31;
    const int wave = (blockIdx.x * blockDim.x + threadIdx.x) >> 5;
    constexpr int nTiles = F >> 4;
    constexpr int totalTiles = (N_NODES >> 4) * nTiles;
    if (wave >= totalTiles) return;               // wave-uniform: EXEC stays all-ones
    const int mTile = wave / nTiles;
    const int nTile = wave - mTile * nTiles;
    const int mrow = (mTile << 4) + (lane & 15);  // A row this lane feeds
    const int ncol = (nTile << 4) + (lane & 15);  // B/C column this lane feeds
    const int kh   = (lane >> 4) << 1;            // 0 (lanes 0-15) or 2 (lanes 16-31)

    const float* __restrict__ Xr = X + (size_t)mrow * K + kh;   // 8B-aligned
    const float* __restrict__ Wc = W + ncol + (size_t)kh * F;

    v8f acc = {};
#pragma unroll 8
    for (int kb = 0; kb < K; kb += 4) {
        v2f a = *(const v2f*)(Xr + kb);           // global_load_b64
        v2f b;
        b.x = Wc[(size_t)kb * F];
        b.y = Wc[(size_t)(kb + 1) * F];
        acc = __builtin_amdgcn_wmma_f32_16x16x4_f32(false, a, false, b,
                                                    (short)0, acc, false, false);
    }
    const int mBase = (mTile << 4) + ((lane >> 4) << 3);  // rows 0-7 or 8-15
#pragma unroll
    for (int r = 0; r < 8; ++r)
        Y[(size_t)(mBase + r) * F + ncol] = acc[r];
}

// ---------------- self-loop term initializes accumulator ----------------
__global__ void k_selfloop(const float* __restrict__ Y, const float* __restrict__ dinv,
                           float* __restrict__ A, int F) {
    const int chunks = F >> 2;
    int i = blockIdx.x * blockDim.x + threadIdx.x;
    if (i >= N_NODES * chunks) return;
    int n = i / chunks, c = i - n * chunks;
    float s = dinv[n]; s *= s;
    float4 v = ((const float4*)(Y + (size_t)n * F))[c];
    float4 o; o.x = v.x * s; o.y = v.y * s; o.z = v.z * s; o.w = v.w * s;
    ((float4*)(A + (size_t)n * F))[c] = o;
}

// ---------------- edge scatter: A[dst] += dinv[s]*dinv[d] * Y[src] ----------------
__global__ void k_edge_scatter(const int* __restrict__ src, const int* __restrict__ dst,
                               const float* __restrict__ dinv, const float* __restrict__ Y,
                               float* __restrict__ A, int F, int logChunks) {
    int i = blockIdx.x * blockDim.x + threadIdx.x;
    const int chunks = 1 << logChunks;            // F/4
    if (i >= (N_EDGES << logChunks)) return;
    const int c = i & (chunks - 1);
    const int e = i >> logChunks;
    const int s = src[e], d = dst[e];
    const float nrm = dinv[s] * dinv[d];
    float4 v = ((const float4*)(Y + (size_t)s * F))[c];
    float* o = A + (size_t)d * F + (c << 2);
    atomAddF(o + 0, nrm * v.x);
    atomAddF(o + 1, nrm * v.y);
    atomAddF(o + 2, nrm * v.z);
    atomAddF(o + 3, nrm * v.w);
}

// ---------------- bias + optional ReLU (F is a power of two) ----------------
__global__ void k_bias_act(float* __restrict__ A, const float* __restrict__ b,
                           int F, int relu) {
    int i = blockIdx.x * blockDim.x + threadIdx.x;
    if (i >= N_NODES * F) return;
    float v = A[i] + b[i & (F - 1)];
    if (relu) v = fmaxf(v, 0.0f);
    A[i] = v;
}

// ---------------- pooling ----------------
__global__ void k_zero_pool(float* __restrict__ pooled, unsigned* __restrict__ cnt) {
    int i = blockIdx.x * blockDim.x + threadIdx.x;
    if (i < N_GRAPHS * H3) pooled[i] = 0.0f;
    else if (i < N_GRAPHS * H3 + N_GRAPHS) cnt[i - N_GRAPHS * H3] = 0u;
}
__global__ void k_pool(const int* __restrict__ batch, const float* __restrict__ A,
                       float* __restrict__ pooled, unsigned* __restrict__ cnt) {
    int i = blockIdx.x * blockDim.x + threadIdx.x;   // node * 16 float4-chunks
    if (i >= N_NODES * 16) return;
    const int n = i >> 4, c = i & 15;
    const int g = batch[n];
    float4 v = ((const float4*)(A + (size_t)n * H3))[c];
    float* o = pooled + (size_t)g * H3 + (c << 2);
    atomAddF(o + 0, v.x); atomAddF(o + 1, v.y);
    atomAddF(o + 2, v.z); atomAddF(o + 3, v.w);
    if (c == 0) atomicAdd(&cnt[g], 1u);
}

// ---------------- head: logits, sigmoid, BCE loss ----------------
__global__ void k_head(const float* __restrict__ pooled, const unsigned* __restrict__ cnt,
                       const float* __restrict__ Wl, const float* __restrict__ bl,
                       const int* __restrict__ targets, float* __restrict__ out) {
    const int g = threadIdx.x;                     // 512 threads, 1 block
    const float inv = 1.0f / fmaxf((float)cnt[g], 1.0f);
    float z = bl[0];
    for (int f = 0; f < H3; ++f) z += pooled[(size_t)g * H3 + f] * inv * Wl[f];
    out[g] = 1.0f / (1.0f + expf(-z));
    const float y = (float)targets[g];
    float l = fmaxf(z, 0.0f) + log1pf(expf(-fabsf(z))) - z * y;
    __shared__ float sm[N_GRAPHS];
    sm[g] = l;
    __syncthreads();
    for (int s = N_GRAPHS >> 1; s > 0; s >>= 1) {
        if (g < s) sm[g] += sm[g + s];
        __syncthreads();
    }
    if (g == 0) out[N_GRAPHS] = sm[0] / (float)N_GRAPHS;
}

// ---------------- driver ----------------
extern "C" void kernel_launch(void* const* d_in, const int* in_sizes, int n_in,
                              void* d_out, int out_size, void* d_ws, size_t ws_size,
                              hipStream_t stream) {
    const float* x  = (const float*)d_in[0];
    const int* ei   = (const int*)d_in[1];
    const int* src  = ei;
    const int* dst  = ei + N_EDGES;
    const int* batch   = (const int*)d_in[2];
    const int* targets = (const int*)d_in[3];
    const float* W1 = (const float*)d_in[4];  const float* b1 = (const float*)d_in[5];
    const float* W2 = (const float*)d_in[6];  const float* b2 = (const float*)d_in[7];
    const float* W3 = (const float*)d_in[8];  const float* b3 = (const float*)d_in[9];
    const float* Wl = (const float*)d_in[10]; const float* bl = (const float*)d_in[11];
    float* out = (float*)d_out;

    char* ws = (char*)d_ws;
    const size_t BIG  = (size_t)N_NODES * H1 * sizeof(float);   // 51.2 MB
    const size_t XPAD = (size_t)N_NODES * KPAD * sizeof(float); // 6.4 MB
    float*    buf0   = (float*)ws;                              // GEMM outputs
    float*    buf1   = (float*)(ws + BIG);                      // aggregates / next input
    float*    xpad   = (float*)(ws + 2 * BIG);
    unsigned* deg    = (unsigned*)(ws + 2 * BIG + XPAD);
    float*    dinv   = (float*)(ws + 2 * BIG + XPAD + (size_t)N_NODES * 4);
    float*    pooled = (float*)(ws + 2 * BIG + XPAD + (size_t)N_NODES * 8);
    unsigned* cnt    = (unsigned*)(ws + 2 * BIG + XPAD + (size_t)N_NODES * 8
                                   + (size_t)N_GRAPHS * H3 * 4);

    const int T = 256;
    auto blk = [](int n, int t) { return (n + t - 1) / t; };

    // degrees & symmetric normalization; pad x to K=32
    k_deg_init <<<blk(N_NODES, T), T, 0, stream>>>(deg);
    k_deg_count<<<blk(N_EDGES, T), T, 0, stream>>>(dst, deg);
    k_deg_inv  <<<blk(N_NODES, T), T, 0, stream>>>(deg, dinv);
    k_pack_x   <<<blk(N_NODES * KPAD, T), T, 0, stream>>>(x, xpad);

    auto aggregate = [&](const float* b, int F, int logChunks, int relu) {
        k_selfloop    <<<blk(N_NODES * (F >> 2), T), T, 0, stream>>>(buf0, dinv, buf1, F);
        k_edge_scatter<<<blk(N_EDGES << logChunks, T), T, 0, stream>>>(src, dst, dinv,
                                                                       buf0, buf1, F, logChunks);
        k_bias_act    <<<blk(N_NODES * F, T), T, 0, stream>>>(buf1, b, F, relu);
    };
    auto gblk = [&](int F) { return blk((N_NODES >> 4) * (F >> 4) * 32, T); };

    // layer 1: 30(->32) -> 256, ReLU
    k_gemm_wmma<KPAD, H1><<<gblk(H1), T, 0, stream>>>(xpad, W1, buf0);
    aggregate(b1, H1, 6, 1);
    // layer 2: 256 -> 128, ReLU
    k_gemm_wmma<H1, H2><<<gblk(H2), T, 0, stream>>>(buf1, W2, buf0);
    aggregate(b2, H2, 5, 1);
    // layer 3: 128 -> 64
    k_gemm_wmma<H2, H3><<<gblk(H3), T, 0, stream>>>(buf1, W3, buf0);
    aggregate(b3, H3, 4, 0);

    // global mean pool + head
    k_zero_pool<<<blk(N_GRAPHS * H3 + N_GRAPHS, T), T, 0, stream>>>(pooled, cnt);
    k_pool     <<<blk(N_NODES * 16, T), T, 0, stream>>>(batch, buf1, pooled, cnt);
    k_head     <<<1, N_GRAPHS, 0, stream>>>(pooled, cnt, Wl, bl, targets, out);
}